// CriticNetwork_37804302139539
// MI455X (gfx1250) — compile-verified
//
#include <hip/hip_runtime.h>
#include <hip/hip_bf16.h>

typedef __attribute__((ext_vector_type(2))) float v2f;
typedef __attribute__((ext_vector_type(8))) float v8f;

// ---------------------------------------------------------------------------
// small utility kernels
// ---------------------------------------------------------------------------
__global__ void zero_f32(float* __restrict__ p, long long n) {
    long long i = (long long)blockIdx.x * blockDim.x + threadIdx.x;
    if (i < n) p[i] = 0.0f;
}

__global__ void deg_init(float* __restrict__ deg, int n) {
    int i = blockIdx.x * blockDim.x + threadIdx.x;
    if (i < n) deg[i] = 1.0f;  // self-loop
}

__global__ void deg_count(const int* __restrict__ dst, float* __restrict__ deg, int E) {
    int e = blockIdx.x * blockDim.x + threadIdx.x;
    if (e < E) atomicAdd(&deg[dst[e]], 1.0f);
}

__global__ void deg_rsqrt(float* __restrict__ dinv, int n) {
    int i = blockIdx.x * blockDim.x + threadIdx.x;
    if (i < n) dinv[i] = rsqrtf(dinv[i]);
}

// ---------------------------------------------------------------------------
// fp32 WMMA GEMM:  H[nrows x ND] = X[nrows x KD] @ W[KD x ND]  (+bias, ReLU)
// 256 threads = 8 waves; each wave computes one 16x16 output tile with
// V_WMMA_F32_16X16X4_F32, K stepped by 4. W is staged in LDS.
// ---------------------------------------------------------------------------
template <int KD, int ND, bool EPI>
__global__ __launch_bounds__(256) void wmma_gemm(const float* __restrict__ X,
                                                 const float* __restrict__ W,
                                                 const float* __restrict__ bias,
                                                 float* __restrict__ H, int nrows) {
    constexpr int NT = ND / 16;   // N tiles per block row (2 or 4)
    constexpr int MT = 8 / NT;    // M tiles per block (4 or 2)
    constexpr int ROWS = MT * 16; // rows per block

    __shared__ float sW[KD * ND];
    for (int i = threadIdx.x; i < KD * ND; i += 256) sW[i] = W[i];
    __syncthreads();

    const int wave = threadIdx.x >> 5;
    const int lane = threadIdx.x & 31;
    const int half = lane >> 4;   // 0: lanes 0-15, 1: lanes 16-31
    const int nl   = lane & 15;
    const int m0 = blockIdx.x * ROWS + (wave / NT) * 16;
    const int n0 = (wave % NT) * 16;

    // A-matrix 16x4 f32: lane (0-15) holds row M=lane, VGPR0=K0,VGPR1=K1;
    // lanes 16-31 hold K2,K3.  Row clamped so EXEC stays all-ones for WMMA.
    int row = m0 + nl;
    if (row > nrows - 1) row = nrows - 1;
    const float* xr = X + (long long)row * KD;

    v8f acc = {};
    for (int k0 = 0; k0 < KD; k0 += 4) {
        const int ka = k0 + 2 * half;
        v2f a = {xr[ka], xr[ka + 1]};
        // B-matrix 4x16: VGPR r, half h -> K = r + 2*h, N = lane&15
        v2f b = {sW[ka * ND + n0 + nl], sW[(ka + 1) * ND + n0 + nl]};
        acc = __builtin_amdgcn_wmma_f32_16x16x4_f32(false, a, false, b, (short)0,
                                                    acc, false, false);
    }

    // C/D 16x16 f32: VGPR r holds M = r + 8*half, N = lane&15
#pragma unroll
    for (int r = 0; r < 8; ++r) {
        int m = m0 + r + 8 * half;
        if (m < nrows) {
            float v = acc[r];
            if (EPI) {
                v += bias[n0 + nl];
                v = v > 0.0f ? v : 0.0f;
            }
            H[(long long)m * ND + n0 + nl] = v;
        }
    }
}

// ---------------------------------------------------------------------------
// edge aggregation: agg[dst] += h[src] * dinv[src]*dinv[dst]
// F consecutive lanes handle one edge -> coalesced gather + coalesced atomics.
// h1/h2 fit in the 192MB L2, so this phase stays on-chip.
// ---------------------------------------------------------------------------
template <int F>
__global__ __launch_bounds__(256) void edge_agg(const float* __restrict__ h,
                                                const int* __restrict__ src,
                                                const int* __restrict__ dst,
                                                const float* __restrict__ dinv,
                                                float* __restrict__ agg, int E) {
    constexpr int EPB = 256 / F;
    const int e = blockIdx.x * EPB + threadIdx.x / F;
    const int f = threadIdx.x & (F - 1);
    if (e >= E) return;
    const int s = src[e];
    const int d = dst[e];
    const float coef = dinv[s] * dinv[d];
    atomicAdd(&agg[d * F + f], h[s * F + f] * coef);
}

// z = relu(agg + h*dinv^2 + b), written in place over agg
template <int F>
__global__ void self_bias_relu(float* __restrict__ agg, const float* __restrict__ h,
                               const float* __restrict__ dinv,
                               const float* __restrict__ b, int n) {
    long long t = (long long)blockIdx.x * blockDim.x + threadIdx.x;
    if (t >= (long long)n * F) return;
    const int i = (int)(t / F);
    const int f = (int)(t % F);
    const float di = dinv[i];
    float v = agg[t] + h[t] * di * di + b[f];
    agg[t] = v > 0.0f ? v : 0.0f;
}

// ---------------------------------------------------------------------------
// head: v[i] = h3[i,:] . Wm2 + bm2 ; out[g] = mean over nodes of graph g
// one wave per node, shfl reduction, atomic mean accumulation.
// ---------------------------------------------------------------------------
__global__ __launch_bounds__(256) void value_reduce(const float* __restrict__ h3,
                                                    const float* __restrict__ Wm2,
                                                    const float* __restrict__ bm2,
                                                    const int* __restrict__ nn_ptr,
                                                    float* __restrict__ out, int nrows) {
    const int node = (int)(((long long)blockIdx.x * blockDim.x + threadIdx.x) >> 5);
    const int lane = threadIdx.x & 31;
    if (node >= nrows) return;
    const float* hr = h3 + (long long)node * 64;
    float p = hr[lane] * Wm2[lane] + hr[lane + 32] * Wm2[lane + 32];
    for (int off = 16; off > 0; off >>= 1) p += __shfl_down(p, off, 32);
    if (lane == 0) {
        const int nn = *nn_ptr;
        const float v = p + bm2[0];
        atomicAdd(&out[node / nn], v / (float)nn);
    }
}

// ---------------------------------------------------------------------------
extern "C" void kernel_launch(void* const* d_in, const int* in_sizes, int n_in,
                              void* d_out, int out_size, void* d_ws, size_t ws_size,
                              hipStream_t stream) {
    const float* x   = (const float*)d_in[0];
    const float* W1c = (const float*)d_in[1];
    const float* b1c = (const float*)d_in[2];
    const float* W2c = (const float*)d_in[3];
    const float* b2c = (const float*)d_in[4];
    const float* Wm1 = (const float*)d_in[5];
    const float* bm1 = (const float*)d_in[6];
    const float* Wm2 = (const float*)d_in[7];
    const float* bm2 = (const float*)d_in[8];
    const int*   ei  = (const int*)d_in[9];
    const int*   nn  = (const int*)d_in[10];

    const int N = in_sizes[0] / 128;   // nodes
    const int E = in_sizes[9] / 2;     // edges
    const int* src = ei;
    const int* dst = ei + E;

    float* out = (float*)d_out;
    float* ws  = (float*)d_ws;

    // workspace layout (floats)
    float* dinv = ws;                       // N
    float* h1   = ws + (long long)N;        // 32N
    float* agg1 = h1 + (long long)N * 32;   // 32N   (becomes z1)
    float* h2   = agg1 + (long long)N * 32; // 64N
    float* agg2 = h2 + (long long)N * 64;   // 64N   (becomes z2)
    float* h3   = h1;                       // 64N, reuses h1+agg1 region

    const int B = 256;
    auto cdiv = [](long long a, long long b) { return (unsigned)((a + b - 1) / b); };

    // output must be zeroed every call (harness poisons once, never restores)
    zero_f32<<<cdiv(out_size, B), B, 0, stream>>>(out, out_size);

    // symmetric-norm coefficients
    deg_init<<<cdiv(N, B), B, 0, stream>>>(dinv, N);
    deg_count<<<cdiv(E, B), B, 0, stream>>>(dst, dinv, E);
    deg_rsqrt<<<cdiv(N, B), B, 0, stream>>>(dinv, N);

    // ---- GCN layer 1: h1 = x @ W1c -------------------------------------
    wmma_gemm<128, 32, false><<<cdiv(N, 64), B, 0, stream>>>(x, W1c, nullptr, h1, N);
    zero_f32<<<cdiv((long long)N * 32, B), B, 0, stream>>>(agg1, (long long)N * 32);
    edge_agg<32><<<cdiv(E, 8), B, 0, stream>>>(h1, src, dst, dinv, agg1, E);
    self_bias_relu<32><<<cdiv((long long)N * 32, B), B, 0, stream>>>(agg1, h1, dinv, b1c, N);

    // ---- GCN layer 2: h2 = z1 @ W2c ------------------------------------
    wmma_gemm<32, 64, false><<<cdiv(N, 32), B, 0, stream>>>(agg1, W2c, nullptr, h2, N);
    zero_f32<<<cdiv((long long)N * 64, B), B, 0, stream>>>(agg2, (long long)N * 64);
    edge_agg<64><<<cdiv(E, 4), B, 0, stream>>>(h2, src, dst, dinv, agg2, E);
    self_bias_relu<64><<<cdiv((long long)N * 64, B), B, 0, stream>>>(agg2, h2, dinv, b2c, N);

    // ---- MLP head: h3 = relu(z2 @ Wm1 + bm1) ---------------------------
    wmma_gemm<64, 64, true><<<cdiv(N, 32), B, 0, stream>>>(agg2, Wm1, bm1, h3, N);

    // ---- v = h3 . Wm2 + bm2 ; per-graph mean ---------------------------
    value_reduce<<<cdiv((long long)N * 32, B), B, 0, stream>>>(h3, Wm2, bm2, nn, out, N);
}